// Model_39676907882504
// MI455X (gfx1250) — compile-verified
//
#include <hip/hip_runtime.h>

// Reference reduces to: out[i, j, 0, :] = value[0, j, 0, :] for all i in [0,S)
//   (softmax over a singleton axis == 1.0 exactly; weight 1.0 applied in f32
//    FMA is bitwise identity). q/k/mask are mathematically dead.
// => replicate flattened value (S*D f32 = 512 KB) into 2048 rows = 1 GiB out.
// Store-bandwidth bound: ~46 us at 23.3 TB/s.
//
// WMMA use: V_WMMA_F32_16X16X4_F32 applies the 1.0 attention weight:
//   A[m,k] = (k==0) ? 1 : 0,  B[0,n] = value[j0+n], B[k>0,*] = 0, C = 0
//   => D[m,n] = value[j0+n]  (exact f32)
// D's C/D VGPR layout (8 VGPRs; lanes 0-15 -> M=r, lanes 16-31 -> M=r+8,
// N = lane&15) feeds a 16x16 output tile per store-loop iteration with every
// lane writing a unique address. Non-temporal stores keep the 1 GiB stream
// out of the 192 MB L2 (value itself stays resident).

typedef float v2f __attribute__((ext_vector_type(2)));
typedef float v8f __attribute__((ext_vector_type(8)));

namespace {
constexpr int S   = 2048;
constexpr int D   = 64;
constexpr int ROW = S * D;                 // 131072 f32 per output row
constexpr int COLS_PER_WAVE  = 16;         // one WMMA tile width
constexpr int WAVES_TOTAL    = ROW / COLS_PER_WAVE;   // 8192
constexpr int WAVES_PER_BLK  = 8;          // 256 threads = 8 wave32
constexpr int BLOCKS         = WAVES_TOTAL / WAVES_PER_BLK; // 1024
constexpr int ROW_TILES      = S / 16;     // 128 row-tiles of 16 output rows
}

__global__ __launch_bounds__(256) void attn_bcast_wmma(
    const float* __restrict__ value, float* __restrict__ out) {
  const int lane = threadIdx.x & 31;
  const int wave = threadIdx.x >> 5;
  const int gw   = blockIdx.x * WAVES_PER_BLK + wave;   // global wave id
  const int j0   = gw * COLS_PER_WAVE;                  // column segment

  // B (4x16 f32): VGPR slot (x, lanes 0-15) holds row K=0, N=lane.
  // All other K-row slots are zero, so only the K=0 row contributes.
  v2f B;
  B.x = (lane < 16) ? value[j0 + lane] : 0.0f;
  B.y = 0.0f;

  // A (16x4 f32): ones in the K=0 column (same (x, lanes 0-15) K-slot as B's
  // data row), zero elsewhere -> D[m,n] = 1.0 * B[0,n].
  v2f A;
  A.x = (lane < 16) ? 1.0f : 0.0f;
  A.y = 0.0f;

  v8f C = {};  // accumulate into zero
  // 8 args: (neg_a, A, neg_b, B, c_mod, C, reuse_a, reuse_b)
  v8f Dt = __builtin_amdgcn_wmma_f32_16x16x4_f32(
      /*neg_a=*/false, A, /*neg_b=*/false, B,
      /*c_mod=*/(short)0, C, /*reuse_a=*/false, /*reuse_b=*/false);
  // Dt[r] == value[j0 + (lane&15)] for output rows m = r + (lane<16 ? 0 : 8)

  const int    n     = lane & 15;
  const size_t mhalf = (lane >> 4) * 8;            // lanes 16-31 -> rows +8
  float* base = out + mhalf * (size_t)ROW + (size_t)j0 + (size_t)n;

  for (int it = 0; it < ROW_TILES; ++it) {
    float* p = base + (size_t)it * 16u * (size_t)ROW;
#pragma unroll
    for (int r = 0; r < 8; ++r) {
      // rows m=r (lanes 0-15) and m=r+8 (lanes 16-31): 32 unique addresses
      __builtin_nontemporal_store(Dt[r], p + (size_t)r * (size_t)ROW);
    }
  }
}

extern "C" void kernel_launch(void* const* d_in, const int* in_sizes, int n_in,
                              void* d_out, int out_size, void* d_ws, size_t ws_size,
                              hipStream_t stream) {
  // setup_inputs order: 0=query, 1=key, 2=value, 3=attn_mask (q/k/mask unused:
  // softmax over a singleton axis is identically 1).
  const float* value = (const float*)d_in[2];
  float* out = (float*)d_out;
  attn_bcast_wmma<<<BLOCKS, 256, 0, stream>>>(value, out);
  (void)d_in; (void)in_sizes; (void)n_in; (void)out_size; (void)d_ws; (void)ws_size;
}